// BatchTaskAlignedAssigner_13091060318509
// MI455X (gfx1250) — compile-verified
//
#include <hip/hip_runtime.h>
#include <hip/hip_bf16.h>

typedef float v2f __attribute__((ext_vector_type(2)));
typedef float v8f __attribute__((ext_vector_type(8)));

#define NUM_CLASSES 80
#define TOPK 13
#define EPSF 1e-7f
#define NB 32
#define NP 8400
#define NG 40
#define GT_TILES 3           // ceil(40/16)
#define P_TILES (NP / 16)    // 525
#define K2_THREADS 256
#define K2_ITEMS 33          // ceil(8400/256)

// ---------------------------------------------------------------------------
// Kernel 1: per-wave 16x16 tile of (g,p). bbox_scores via exact one-hot WMMA
// (f32 16x16x4), then CIoU / is_in_gts / metrics per element.
// ---------------------------------------------------------------------------
__global__ void k1_metrics(const float* __restrict__ pred_bboxes,   // B,P,4
                           const float* __restrict__ pred_scores,   // B,P,80
                           const float* __restrict__ priors,        // P,4
                           const int*   __restrict__ gt_labels,     // B,G
                           const float* __restrict__ gt_bboxes,     // B,G,4
                           float*        __restrict__ overlaps,     // B,G,P
                           float*        __restrict__ metrics,      // B,G,P
                           unsigned char* __restrict__ in_gts)      // B,G,P
{
    const int wavesPerBlock = blockDim.x >> 5;
    int wave = blockIdx.x * wavesPerBlock + (threadIdx.x >> 5);
    int lane = threadIdx.x & 31;
    if (wave >= NB * GT_TILES * P_TILES) return;   // wave-uniform

    int b     = wave / (GT_TILES * P_TILES);
    int rem   = wave % (GT_TILES * P_TILES);
    int gtile = rem / P_TILES;
    int ptile = rem % P_TILES;

    int p  = ptile * 16 + (lane & 15);
    int hi = lane >> 4;                     // which K/M half this lane serves

    // A-matrix (one-hot of labels): lane L holds row M = L%16, K = (L/16)*2 +{0,1}
    int gA   = gtile * 16 + (lane & 15);
    int labA = (gA < NG) ? gt_labels[b * NG + gA] : -1;

    // B-matrix: lane L holds col N = L%16 (prior p), K = (L/16)*2 +{0,1}
    const float* sc = pred_scores + ((size_t)b * NP + p) * NUM_CLASSES;

    v8f acc = {0.f, 0.f, 0.f, 0.f, 0.f, 0.f, 0.f, 0.f};
#pragma unroll
    for (int k = 0; k < NUM_CLASSES / 4; ++k) {
        int K0 = k * 4 + hi * 2;
        v2f a, bfrag;
        a.x = (labA == K0)     ? 1.f : 0.f;
        a.y = (labA == K0 + 1) ? 1.f : 0.f;
        bfrag.x = sc[K0];
        bfrag.y = sc[K0 + 1];
        acc = __builtin_amdgcn_wmma_f32_16x16x4_f32(
            false, a, false, bfrag, (short)0, acc, false, false);
    }
    // acc[r]: bbox_score for g = gtile*16 + r + 8*hi, prior p (lane-constant)

    const float4 pb = *reinterpret_cast<const float4*>(pred_bboxes + ((size_t)b * NP + p) * 4);
    float ptx = priors[p * 4 + 0];
    float pty = priors[p * 4 + 1];
    float w1 = pb.z - pb.x, h1 = pb.w - pb.y;
    float atan1 = atanf(w1 / (h1 + EPSF));
    const float C4PI2 = 0.40528473456935109f;   // 4/pi^2

#pragma unroll
    for (int rr = 0; rr < 8; ++rr) {
        int g = gtile * 16 + rr + 8 * hi;
        if (g >= NG) continue;
        const float4 tb = *reinterpret_cast<const float4*>(gt_bboxes + ((size_t)b * NG + g) * 4);

        // ---- CIoU (clipped at 0, matching reference) ----
        float ow = fmaxf(fminf(pb.z, tb.z) - fmaxf(pb.x, tb.x), 0.f);
        float oh = fmaxf(fminf(pb.w, tb.w) - fmaxf(pb.y, tb.y), 0.f);
        float overlap = ow * oh;
        float w2 = tb.z - tb.x, h2 = tb.w - tb.y;
        float uni  = w1 * h1 + w2 * h2 - overlap + EPSF;
        float ious = overlap / uni;
        float ew = fmaxf(fmaxf(pb.z, tb.z) - fminf(pb.x, tb.x), 0.f);
        float eh = fmaxf(fmaxf(pb.w, tb.w) - fminf(pb.y, tb.y), 0.f);
        float enclose = ew * ew + eh * eh + EPSF;
        float dx = (tb.x + tb.z) - (pb.x + pb.z);
        float dy = (tb.y + tb.w) - (pb.y + pb.w);
        float rho2 = (dx * dx + dy * dy) * 0.25f;
        float da = atanf(w2 / (h2 + EPSF)) - atan1;
        float wh_ratio = C4PI2 * da * da;
        float alpha = wh_ratio / (wh_ratio - ious + (1.f + EPSF));
        float res = ious - (rho2 / enclose + alpha * wh_ratio);
        res = fminf(fmaxf(res, -1.f), 1.f);
        float ov = fmaxf(res, 0.f);

        float o2 = ov * ov;
        float o6 = o2 * o2 * o2;
        float align = acc[rr] * o6;         // ALPHA=1, BETA=6

        float d0 = ptx - tb.x, d1 = pty - tb.y, d2 = tb.z - ptx, d3 = tb.w - pty;
        float mn = fminf(fminf(d0, d1), fminf(d2, d3));
        float ing = (mn > 1e-9f) ? 1.f : 0.f;

        size_t idx = ((size_t)(b * NG + g)) * NP + p;
        overlaps[idx] = ov;
        metrics[idx]  = align * ing;
        in_gts[idx]   = (unsigned char)ing;
    }
}

// ---------------------------------------------------------------------------
// Kernel 2: per-(b,g) iterative top-13 (jax tie-break: lowest index first),
// rewrite the metrics row in place as pos_mask = in_topk * in_gts * pad.
// ---------------------------------------------------------------------------
__global__ void k2_topk(const unsigned char* __restrict__ in_gts,
                        const float* __restrict__ pad_flag,        // B,G
                        float* __restrict__ mrow)                  // metrics->pos_mask
{
    int bg  = blockIdx.x;
    int tid = threadIdx.x;
    float* row = mrow + (size_t)bg * NP;
    const unsigned char* ing = in_gts + (size_t)bg * NP;
    float pad = pad_flag[bg];

    float vals[K2_ITEMS];
#pragma unroll
    for (int j = 0; j < K2_ITEMS; ++j) {
        int p = tid + j * K2_THREADS;
        vals[j] = (p < NP) ? row[p] : -1.f;
    }

    __shared__ float sv[K2_THREADS];
    __shared__ int   si[K2_THREADS];
    __shared__ int   winners[TOPK];

    for (int t = 0; t < TOPK; ++t) {
        float bv = -2.f; int bi = NP;
#pragma unroll
        for (int j = 0; j < K2_ITEMS; ++j) {
            int p = tid + j * K2_THREADS;
            float v = vals[j];
            if (p < NP && (v > bv || (v == bv && p < bi))) { bv = v; bi = p; }
        }
        sv[tid] = bv; si[tid] = bi;
        __syncthreads();
        for (int s = K2_THREADS / 2; s > 0; s >>= 1) {
            if (tid < s) {
                float ov = sv[tid + s]; int oi = si[tid + s];
                if (ov > sv[tid] || (ov == sv[tid] && oi < si[tid])) { sv[tid] = ov; si[tid] = oi; }
            }
            __syncthreads();
        }
        if (tid == 0) winners[t] = si[0];
        __syncthreads();
        int w = winners[t];
#pragma unroll
        for (int j = 0; j < K2_ITEMS; ++j) {
            if (tid + j * K2_THREADS == w) vals[j] = -1.f;   // remove selected
        }
        __syncthreads();
    }

    // pos_mask: zero row, then scatter the 13 winners (pad<=0 -> dup idx 0 -> all zero)
#pragma unroll
    for (int j = 0; j < K2_ITEMS; ++j) {
        int p = tid + j * K2_THREADS;
        if (p < NP) row[p] = 0.f;
    }
    __syncthreads();
    if (tid < TOPK && pad > 0.f) {
        int w = winners[tid];
        row[w] = ing[w] ? pad : 0.f;
    }
}

// ---------------------------------------------------------------------------
// Kernel 3: per prior — fg count, multi-GT resolution via argmax(overlaps),
// pos_mask update, labels / bboxes / fg outputs.
// ---------------------------------------------------------------------------
__global__ void k3_resolve(const float* __restrict__ overlaps,
                           float* __restrict__ pos_mask,
                           const int* __restrict__ gt_labels,
                           const float* __restrict__ gt_bboxes,
                           float* __restrict__ out_labels,
                           float* __restrict__ out_bboxes,
                           float* __restrict__ out_fg)
{
    int i = blockIdx.x * blockDim.x + threadIdx.x;
    if (i >= NB * NP) return;
    int b = i / NP, p = i % NP;
    size_t base = (size_t)b * NG * NP + p;

    float fg = 0.f;
    for (int g = 0; g < NG; ++g) fg += pos_mask[base + (size_t)g * NP];

    int gt_idx;
    if (fg > 1.f) {
        float bv = -INFINITY; int bgx = 0;
        for (int g = 0; g < NG; ++g) {
            float v = overlaps[base + (size_t)g * NP];
            if (v > bv) { bv = v; bgx = g; }                  // first-max (jnp.argmax)
        }
        for (int g = 0; g < NG; ++g)
            pos_mask[base + (size_t)g * NP] = (g == bgx) ? 1.f : 0.f;
        fg = 1.f;
        gt_idx = bgx;
    } else {
        float bv = -INFINITY; int bgx = 0;
        for (int g = 0; g < NG; ++g) {
            float v = pos_mask[base + (size_t)g * NP];
            if (v > bv) { bv = v; bgx = g; }
        }
        gt_idx = bgx;
    }

    int lab = gt_labels[b * NG + gt_idx];
    lab = (lab < 0) ? 0 : lab;
    out_labels[i] = (float)lab;
    const float4 tb = *reinterpret_cast<const float4*>(gt_bboxes + ((size_t)b * NG + gt_idx) * 4);
    *reinterpret_cast<float4*>(out_bboxes + (size_t)i * 4) = tb;
    out_fg[i] = (fg > 0.f) ? 1.f : 0.f;
}

// ---------------------------------------------------------------------------
// Kernel 4: per (b,g): pos_align = max_p(align*pos_mask), pos_ious = max_p(ov*pos_mask)
// ---------------------------------------------------------------------------
__global__ void k4_stats(const float* __restrict__ overlaps,
                         const float* __restrict__ pos_mask,
                         const float* __restrict__ pred_scores,
                         const int* __restrict__ gt_labels,
                         float* __restrict__ pos_align,
                         float* __restrict__ pos_ious)
{
    int bg = blockIdx.x;
    int b  = bg / NG;
    int tid = threadIdx.x;
    int lab = gt_labels[bg];
    lab = (lab < 0) ? 0 : (lab >= NUM_CLASSES ? NUM_CLASSES - 1 : lab);

    const float* ov = overlaps + (size_t)bg * NP;
    const float* pm = pos_mask + (size_t)bg * NP;
    const float* sc = pred_scores + (size_t)b * NP * NUM_CLASSES + lab;

    float ma = 0.f, mi = 0.f;
    for (int p = tid; p < NP; p += blockDim.x) {
        float m = pm[p];
        if (m > 0.f) {
            float o = ov[p];
            float o2 = o * o, o6 = o2 * o2 * o2;
            float s = sc[(size_t)p * NUM_CLASSES];
            ma = fmaxf(ma, s * o6 * m);
            mi = fmaxf(mi, o * m);
        }
    }
    __shared__ float sa[256], sb_[256];
    sa[tid] = ma; sb_[tid] = mi;
    __syncthreads();
    for (int s = 128; s > 0; s >>= 1) {
        if (tid < s) {
            sa[tid]  = fmaxf(sa[tid],  sa[tid + s]);
            sb_[tid] = fmaxf(sb_[tid], sb_[tid + s]);
        }
        __syncthreads();
    }
    if (tid == 0) { pos_align[bg] = sa[0]; pos_ious[bg] = sb_[0]; }
}

// ---------------------------------------------------------------------------
// Kernel 5: per prior — norm = max_g(align*pm*pos_ious/(pos_align+eps)),
// write one-hot assigned_scores * norm * (fg>0).
// ---------------------------------------------------------------------------
__global__ void k5_scores(const float* __restrict__ overlaps,
                          const float* __restrict__ pos_mask,
                          const float* __restrict__ pred_scores,
                          const int* __restrict__ gt_labels,
                          const float* __restrict__ pos_align,
                          const float* __restrict__ pos_ious,
                          const float* __restrict__ out_labels,
                          const float* __restrict__ out_fg,
                          float* __restrict__ out_scores)
{
    int i = blockIdx.x * blockDim.x + threadIdx.x;
    if (i >= NB * NP) return;
    int b = i / NP, p = i % NP;
    size_t base = (size_t)b * NG * NP + p;

    float norm = 0.f;
    for (int g = 0; g < NG; ++g) {
        float m = pos_mask[base + (size_t)g * NP];
        if (m > 0.f) {
            float o = overlaps[base + (size_t)g * NP];
            float o2 = o * o, o6 = o2 * o2 * o2;
            int lab = gt_labels[b * NG + g];
            lab = (lab < 0) ? 0 : (lab >= NUM_CLASSES ? NUM_CLASSES - 1 : lab);
            float s = pred_scores[((size_t)b * NP + p) * NUM_CLASSES + lab];
            float v = s * o6 * m * pos_ious[b * NG + g] / (pos_align[b * NG + g] + EPSF);
            norm = fmaxf(norm, v);
        }
    }
    float fg  = out_fg[i];
    int   lab = (int)out_labels[i];
    float val = (fg > 0.f) ? norm : 0.f;
    float* dst = out_scores + (size_t)i * NUM_CLASSES;
#pragma unroll
    for (int c = 0; c < NUM_CLASSES; ++c) dst[c] = (c == lab) ? val : 0.f;
}

// ---------------------------------------------------------------------------
extern "C" void kernel_launch(void* const* d_in, const int* in_sizes, int n_in,
                              void* d_out, int out_size, void* d_ws, size_t ws_size,
                              hipStream_t stream)
{
    const float* pred_bboxes = (const float*)d_in[0];
    const float* pred_scores = (const float*)d_in[1];
    const float* priors      = (const float*)d_in[2];
    const int*   gt_labels   = (const int*)d_in[3];
    const float* gt_bboxes   = (const float*)d_in[4];
    const float* pad_flag    = (const float*)d_in[5];

    float* out        = (float*)d_out;
    float* out_labels = out;
    float* out_bboxes = out_labels + (size_t)NB * NP;
    float* out_scores = out_bboxes + (size_t)NB * NP * 4;
    float* out_fg     = out_scores + (size_t)NB * NP * NUM_CLASSES;

    const size_t BGP = (size_t)NB * NG * NP;
    float* overlaps = (float*)d_ws;
    float* metrics  = overlaps + BGP;                 // reused as pos_mask after k2
    unsigned char* in_gts = (unsigned char*)(metrics + BGP);
    float* pos_align = (float*)(in_gts + BGP);
    float* pos_ious  = pos_align + NB * NG;

    const int waves     = NB * GT_TILES * P_TILES;    // 50400
    const int k1_blocks = waves / 8;                  // 8 waves (256 thr) per block
    const int np_blocks = (NB * NP + 255) / 256;

    k1_metrics<<<dim3(k1_blocks), dim3(256), 0, stream>>>(
        pred_bboxes, pred_scores, priors, gt_labels, gt_bboxes,
        overlaps, metrics, in_gts);

    k2_topk<<<dim3(NB * NG), dim3(K2_THREADS), 0, stream>>>(
        in_gts, pad_flag, metrics);

    k3_resolve<<<dim3(np_blocks), dim3(256), 0, stream>>>(
        overlaps, metrics, gt_labels, gt_bboxes,
        out_labels, out_bboxes, out_fg);

    k4_stats<<<dim3(NB * NG), dim3(256), 0, stream>>>(
        overlaps, metrics, pred_scores, gt_labels, pos_align, pos_ious);

    k5_scores<<<dim3(np_blocks), dim3(256), 0, stream>>>(
        overlaps, metrics, pred_scores, gt_labels, pos_align, pos_ious,
        out_labels, out_fg, out_scores);
}